// ReducedRankModel_42442866819113
// MI455X (gfx1250) — compile-verified
//
#include <hip/hip_runtime.h>
#include <stdint.h>

// Problem constants (from reference)
#define CC 2048   // neurons
#define TT 1024   // time bins
#define RR 32     // rank
#define KK 128    // trials

// Kernel-2 tiling
#define NSEG   16            // C split into 16 segments -> partials P[s,k,t]
#define SEG_C  (CC / NSEG)   // 128 c-rows per block
#define TILE_C 16            // TDM sub-tile rows staged in LDS (double-buffered)
#define TILE_T 256           // t extent per block (64 threads x float4)
#define NSUB   (SEG_C / TILE_C)

typedef float    f32x4 __attribute__((ext_vector_type(4)));
typedef float    v2f   __attribute__((ext_vector_type(2)));
typedef float    v8f   __attribute__((ext_vector_type(8)));
typedef uint32_t u32x4 __attribute__((ext_vector_type(4)));
typedef int32_t  i32x4 __attribute__((ext_vector_type(4)));
typedef int32_t  i32x8 __attribute__((ext_vector_type(8)));

// ---------------------------------------------------------------------------
// Kernel 1: uv[c,t] = sum_r U[c,t,r] * V[r,t]
// U (256 MB) streamed once with non-temporal 128-bit loads (keep L2 for uv);
// V (128 KB) stays L2-resident; uv (8 MB) stored RT so it stays in L2 for
// kernel 2's 128x re-reads.
// ---------------------------------------------------------------------------
__global__ __launch_bounds__(256) void k1_uv(const float* __restrict__ U,
                                             const float* __restrict__ V,
                                             float* __restrict__ uv) {
  size_t i = (size_t)blockIdx.x * 256u + threadIdx.x;  // linear over C*T
  int t = (int)(i & (TT - 1));                         // lanes -> consecutive t
  const float* up = U + i * RR;                        // 32 contiguous f32 per thread
  float acc = 0.0f;
#pragma unroll
  for (int r = 0; r < RR; r += 4) {
    f32x4 u4 = __builtin_nontemporal_load((const f32x4*)(up + r));
    acc += u4.x * V[(r + 0) * TT + t];
    acc += u4.y * V[(r + 1) * TT + t];
    acc += u4.z * V[(r + 2) * TT + t];
    acc += u4.w * V[(r + 3) * TT + t];
  }
  uv[i] = acc;
}

// ---------------------------------------------------------------------------
// TDM: DMA one (TILE_C x TILE_T) f32 tile of uv (row stride TT) into LDS.
// D# per CDNA5 ISA. Issued by wave 0 only; completion
// tracked with TENSORcnt.
// ---------------------------------------------------------------------------
__device__ __forceinline__ void tdm_load_uv_tile(const float* gsrc, uint32_t lds_byte_off) {
  uint64_t ga = (uint64_t)(uintptr_t)gsrc;
  u32x4 g0;
  g0[0] = 1u;                                                 // count=1, user mode
  g0[1] = lds_byte_off;                                       // lds_addr
  g0[2] = (uint32_t)ga;                                       // global_addr[31:0]
  g0[3] = (uint32_t)((ga >> 32) & 0x1FFFFFFu) | (2u << 30);   // addr[56:32] | type=2
  i32x8 g1;
  g1[0] = (int32_t)(2u << 16);                                // data_size=2 (4 bytes)
  g1[1] = (int32_t)(((uint32_t)TT & 0xFFFFu) << 16);          // tensor_dim0[15:0]
  g1[2] = (int32_t)((((uint32_t)TT) >> 16) | (((uint32_t)CC & 0xFFFFu) << 16)); // dim0 hi | dim1 lo
  g1[3] = (int32_t)((((uint32_t)CC) >> 16) | (((uint32_t)TILE_T) << 16));       // dim1 hi | tile_dim0
  g1[4] = (int32_t)((uint32_t)TILE_C);                        // tile_dim1 (tile_dim2=0)
  g1[5] = (int32_t)((uint32_t)TT);                            // tensor_dim0_stride[31:0]
  g1[6] = 0;
  g1[7] = 0;
  i32x4 gz = {0, 0, 0, 0};                                    // groups 2/3 unused (2D tile)
#if defined(__clang_major__) && __clang_major__ >= 23
  i32x8 gz8 = {0, 0, 0, 0, 0, 0, 0, 0};
  __builtin_amdgcn_tensor_load_to_lds(g0, g1, gz, gz, gz8, 0);
#else
  __builtin_amdgcn_tensor_load_to_lds(g0, g1, gz, gz, 0);
#endif
}

// ---------------------------------------------------------------------------
// Kernel 2: P[s,k,t] = sum_{c in segment s} uv[c,t] * X[k,c,t]
// Grid = K * NSEG * (T/TILE_T) = 8192 blocks of 64 threads (2 waves).
// X (1 GB) streamed with NT 128-bit loads; uv tiles double-buffered into LDS
// by the TDM so the DMA hides under the X stream / FMA loop.
// Pipeline per iteration cc:
//   wait(TDM cc) -> barrier -> issue(TDM cc+1, other buffer) -> compute(cc)
// ---------------------------------------------------------------------------
__global__ __launch_bounds__(64) void k2_partials(const float* __restrict__ X,
                                                  const float* __restrict__ uv,
                                                  float* __restrict__ P) {
  __shared__ __align__(16) float lds_uv[2][TILE_C * TILE_T];  // 2 x 16 KB

  int bid = blockIdx.x;
  int tch = bid & (TT / TILE_T - 1);          // 0..3
  int s   = (bid >> 2) & (NSEG - 1);          // 0..15
  int k   = bid >> 6;                         // 0..127
  int t0  = tch * TILE_T;
  int c0  = s * SEG_C;
  int tid = threadIdx.x;

  uint32_t lds_b0 = (uint32_t)(uintptr_t)&lds_uv[0][0];       // low 32 bits = LDS offset
  uint32_t lds_b1 = (uint32_t)(uintptr_t)&lds_uv[1][0];

  f32x4 acc = {0.0f, 0.0f, 0.0f, 0.0f};

  if (tid < 32) {                                             // wave 0: prime buffer 0
    tdm_load_uv_tile(uv + (size_t)c0 * TT + t0, lds_b0);
  }

#pragma unroll 1
  for (int cc = 0; cc < NSUB; ++cc) {
    if (tid < 32) {
      __builtin_amdgcn_s_wait_tensorcnt(0);                   // tile cc landed in LDS
    }
    __syncthreads();                                          // publish tile cc; prev compute done
    if (cc + 1 < NSUB && tid < 32) {                          // prefetch tile cc+1 (other buffer)
      tdm_load_uv_tile(uv + (size_t)(c0 + (cc + 1) * TILE_C) * TT + t0,
                       ((cc + 1) & 1) ? lds_b1 : lds_b0);
    }

    int cbase = c0 + cc * TILE_C;
    const float* xp = X + ((size_t)k * CC + cbase) * TT + t0 + tid * 4;
    const float* up = &lds_uv[cc & 1][tid * 4];
#pragma unroll
    for (int c = 0; c < TILE_C; ++c) {
      f32x4 xv  = __builtin_nontemporal_load((const f32x4*)(xp + (size_t)c * TT));
      f32x4 uvv = *(const f32x4*)(up + c * TILE_T);
      acc.x += xv.x * uvv.x;
      acc.y += xv.y * uvv.y;
      acc.z += xv.z * uvv.z;
      acc.w += xv.w * uvv.w;
    }
  }

  float* pp = P + ((size_t)s * KK + k) * TT + t0 + tid * 4;
  *(f32x4*)pp = acc;
}

// ---------------------------------------------------------------------------
// Kernel 3: Y[k,t] = sum_{s=0..15} P[s,k,t] via V_WMMA_F32_16X16X4_F32.
// One wave per (k, 16-t tile). A (16x4 f32): lanes 0-15 hold {K=0,K=1},
// lanes 16-31 hold {K=2,K=3} (ISA §7.12.2) -> lane l loads P for segments
// 4q+{0,1} or 4q+{2,3} at t0+(l&15). B = all-ones -> every D column equals
// the s-sum; D layout: lane 0 VGPRs 0..7 = M 0..7, lane 16 = M 8..15,
// so two lanes store the 16 results as 2x b128 each.
// ---------------------------------------------------------------------------
__global__ __launch_bounds__(32) void k3_reduce_wmma(const float* __restrict__ P,
                                                     float* __restrict__ Y) {
  int bid = blockIdx.x;                       // K * (T/16) = 8192
  int t0  = (bid & (TT / 16 - 1)) * 16;
  int k   = bid >> 6;
  int l   = threadIdx.x;
  int m   = l & 15;
  int sOf = (l < 16) ? 0 : 2;

  const size_t KT = (size_t)KK * TT;
  const float* base = P + (size_t)k * TT + t0 + m;

  v8f acc = {0.f, 0.f, 0.f, 0.f, 0.f, 0.f, 0.f, 0.f};
  v2f bones = {1.0f, 1.0f};
#pragma unroll
  for (int q = 0; q < 4; ++q) {
    v2f a;
    a.x = base[(size_t)(4 * q + sOf + 0) * KT];
    a.y = base[(size_t)(4 * q + sOf + 1) * KT];
    acc = __builtin_amdgcn_wmma_f32_16x16x4_f32(false, a, false, bones,
                                                (short)0, acc, false, false);
  }

  f32x4 lo = __builtin_shufflevector(acc, acc, 0, 1, 2, 3);   // M = base..base+3
  f32x4 hi = __builtin_shufflevector(acc, acc, 4, 5, 6, 7);   // M = base+4..base+7
  float* yp = Y + (size_t)k * TT + t0;
  if (l == 0) {                                               // M = 0..7
    *(f32x4*)(yp + 0) = lo;
    *(f32x4*)(yp + 4) = hi;
  }
  if (l == 16) {                                              // M = 8..15
    *(f32x4*)(yp + 8)  = lo;
    *(f32x4*)(yp + 12) = hi;
  }
}

// ---------------------------------------------------------------------------
// Launch: d_in = [recording_index(i32), X(K,C,T), U(C,T,R), V(R,T)], out=(K,T)
// ws: uv (8 MB) at offset 0, P (8 MB) after it.
// ---------------------------------------------------------------------------
extern "C" void kernel_launch(void* const* d_in, const int* in_sizes, int n_in,
                              void* d_out, int out_size, void* d_ws, size_t ws_size,
                              hipStream_t stream) {
  const float* X = (const float*)d_in[1];
  const float* U = (const float*)d_in[2];
  const float* V = (const float*)d_in[3];
  float* Y = (float*)d_out;

  float* uv = (float*)d_ws;
  float* P  = (float*)((char*)d_ws + (size_t)CC * TT * sizeof(float));

  k1_uv<<<(CC * TT) / 256, 256, 0, stream>>>(U, V, uv);
  k2_partials<<<KK * NSEG * (TT / TILE_T), 64, 0, stream>>>(X, uv, P);
  k3_reduce_wmma<<<KK * (TT / 16), 32, 0, stream>>>(P, Y);
}